// LossAutoencoder_23476291240772
// MI455X (gfx1250) — compile-verified
//
#include <hip/hip_runtime.h>
#include <math.h>

// Problem constants (from reference): B=256, N=64, DV=768, DE=384
#define LB   256
#define LN   64
#define LDV  768
#define LDE  384
#define BN   (LB * LN)           // 16384 rows for loss_auto
#define AUTO_BLOCKS (BN / 8)     // 8 rows (one per wave) per 256-thread block = 2048

typedef float v2f __attribute__((ext_vector_type(2)));
typedef float v8f __attribute__((ext_vector_type(8)));

// Wave32-wide sum of 64 partials (2 per lane) using V_WMMA_F32_16X16X4_F32.
// A(16x4) holds the partials (lane m: K=0,1 ; lane m+16: K=2,3), B = ones(4x16),
// so D[m][n] = sum_k A[m][k]. Summing a lane's 8 D VGPRs yields half-wave sums
// (lanes 0-15: rows 0-7 => lanes {0-7,16-23}; lanes 16-31: rows 8-15), and one
// shfl_xor(16) broadcasts the full-wave total to every lane.
// EXEC must be all ones at call sites (it is: straight-line full-block code).
__device__ __forceinline__ float wave_allreduce_wmma(float p0, float p1) {
    v2f a; a.x = p0;   a.y = p1;
    v2f b; b.x = 1.0f; b.y = 1.0f;
    v8f c = {};
    v8f d = __builtin_amdgcn_wmma_f32_16x16x4_f32(
        /*neg_a=*/false, a, /*neg_b=*/false, b,
        /*c_mod=*/(short)0, c, /*reuse_a=*/false, /*reuse_b=*/false);
    float r = d[0] + d[1] + d[2] + d[3] + d[4] + d[5] + d[6] + d[7];
    r += __shfl_xor(r, 16);
    return r;
}

// ---------------- loss_auto: sum over rows of ||o - d||_2 ----------------
// One wave per row of 768 floats: 6 float4 loads per lane per tensor.
__global__ void __launch_bounds__(256) k_auto(const float* __restrict__ o,
                                              const float* __restrict__ dec,
                                              float* __restrict__ partials) {
    const int wave = threadIdx.x >> 5;
    const int lane = threadIdx.x & 31;
    const int row  = blockIdx.x * 8 + wave;
    const size_t base = (size_t)row * LDV + (size_t)lane * 4;
    const float4* op = (const float4*)(o + base);
    const float4* dp = (const float4*)(dec + base);

    float p0 = 0.0f, p1 = 0.0f;
#pragma unroll
    for (int k = 0; k < 6; ++k) {
        float4 a = op[k * 32];          // stride 128 floats between chunks
        float4 b = dp[k * 32];
        float dx = a.x - b.x, dy = a.y - b.y, dz = a.z - b.z, dw = a.w - b.w;
        p0 += dx * dx + dy * dy;
        p1 += dz * dz + dw * dw;
    }
    float ss = wave_allreduce_wmma(p0, p1);

    __shared__ float lds[8];
    if (lane == 0) lds[wave] = sqrtf(ss);
    __syncthreads();
    if (threadIdx.x == 0) {
        float s = 0.0f;
#pragma unroll
        for (int w = 0; w < 8; ++w) s += lds[w];
        partials[blockIdx.x] = s;       // fixed order -> deterministic
    }
}

// ---------------- loss_vi: per batch, ||sum_n x_n/||x_n|| ||^2 ----------------
// One block per batch. Each wave owns 8 rows; a row's 12 values stay in
// registers: compute ||row||^2 via the WMMA wave-reduce, rescale in-register,
// accumulate per-lane partials of s (fixed columns per lane). Single pass
// over global memory; cross-wave combine through 12KB LDS.
__global__ void __launch_bounds__(256) k_vi(const float* __restrict__ x,
                                            float* __restrict__ partials) {
    const int wave = threadIdx.x >> 5;
    const int lane = threadIdx.x & 31;
    const size_t bbase = (size_t)blockIdx.x * LN * LDE;

    float4 s0 = {0,0,0,0}, s1 = {0,0,0,0}, s2 = {0,0,0,0};

#pragma unroll
    for (int i = 0; i < 8; ++i) {
        const int n = wave * 8 + i;
        const float4* xp = (const float4*)(x + bbase + (size_t)n * LDE + (size_t)lane * 4);
        float4 v0 = xp[0];              // cols lane*4      .. +3
        float4 v1 = xp[32];             // cols 128+lane*4  .. +3
        float4 v2 = xp[64];             // cols 256+lane*4  .. +3
        float p0 = v0.x*v0.x + v0.y*v0.y + v0.z*v0.z + v0.w*v0.w
                 + v1.x*v1.x + v1.y*v1.y;
        float p1 = v1.z*v1.z + v1.w*v1.w
                 + v2.x*v2.x + v2.y*v2.y + v2.z*v2.z + v2.w*v2.w;
        float ss  = wave_allreduce_wmma(p0, p1);
        float inv = 1.0f / sqrtf(ss);
        s0.x += v0.x * inv; s0.y += v0.y * inv; s0.z += v0.z * inv; s0.w += v0.w * inv;
        s1.x += v1.x * inv; s1.y += v1.y * inv; s1.z += v1.z * inv; s1.w += v1.w * inv;
        s2.x += v2.x * inv; s2.y += v2.y * inv; s2.z += v2.z * inv; s2.w += v2.w * inv;
    }

    __shared__ __align__(16) float s_lds[8 * LDE];   // 12 KB
    __shared__ float red[256];

    float* dst = &s_lds[wave * LDE + lane * 4];
    *(float4*)(dst      ) = s0;
    *(float4*)(dst + 128) = s1;
    *(float4*)(dst + 256) = s2;
    __syncthreads();

    float acc = 0.0f;
    for (int dcol = threadIdx.x; dcol < LDE; dcol += 256) {
        float s = 0.0f;
#pragma unroll
        for (int w = 0; w < 8; ++w) s += s_lds[w * LDE + dcol];
        acc += s * s;
    }
    red[threadIdx.x] = acc;
    __syncthreads();
    for (int stride = 128; stride > 0; stride >>= 1) {
        if (threadIdx.x < stride) red[threadIdx.x] += red[threadIdx.x + stride];
        __syncthreads();
    }
    if (threadIdx.x == 0) partials[blockIdx.x] = red[0];
}

// ---------------- finalize: fixed-order sums + scalar math ----------------
__global__ void __launch_bounds__(256) k_final(const float* __restrict__ auto_p,
                                               const float* __restrict__ vi_p,
                                               float* __restrict__ out) {
    __shared__ float red[256];
    const int t = threadIdx.x;

    float a = 0.0f;
#pragma unroll
    for (int i = 0; i < AUTO_BLOCKS / 256; ++i) a += auto_p[t + i * 256];
    red[t] = a;
    __syncthreads();
    for (int s = 128; s > 0; s >>= 1) {
        if (t < s) red[t] += red[t + s];
        __syncthreads();
    }
    float loss_auto = red[0];
    __syncthreads();

    red[t] = vi_p[t];
    __syncthreads();
    for (int s = 128; s > 0; s >>= 1) {
        if (t < s) red[t] += red[t + s];
        __syncthreads();
    }

    if (t == 0) {
        float sum_s_sq = red[0];
        const float bn = (float)BN;                   // 16384
        float loss_vi  = 0.5f * (sum_s_sq - bn);
        float auto_n   = loss_auto / bn;
        float vi_n     = loss_vi / (bn * (float)(LN + 1) * 0.5f);  // /532480
        out[0] = auto_n - vi_n;
        out[1] = auto_n;
        out[2] = vi_n;
    }
}

extern "C" void kernel_launch(void* const* d_in, const int* in_sizes, int n_in,
                              void* d_out, int out_size, void* d_ws, size_t ws_size,
                              hipStream_t stream) {
    (void)in_sizes; (void)n_in; (void)out_size; (void)ws_size;
    const float* orig = (const float*)d_in[0];   // (B,N,DV) f32
    const float* deco = (const float*)d_in[1];   // (B,N,DV) f32
    const float* envi = (const float*)d_in[2];   // (B,N,DE) f32
    float* out = (float*)d_out;                  // 3 floats

    float* ws      = (float*)d_ws;
    float* auto_p  = ws;                 // [AUTO_BLOCKS] = 2048 floats
    float* vi_p    = ws + AUTO_BLOCKS;   // [LB] = 256 floats

    k_auto <<<AUTO_BLOCKS, 256, 0, stream>>>(orig, deco, auto_p);
    k_vi   <<<LB,          256, 0, stream>>>(envi, vi_p);
    k_final<<<1,           256, 0, stream>>>(auto_p, vi_p, out);
}